// mimoaf_38431367365130
// MI455X (gfx1250) — compile-verified
//
#include <hip/hip_runtime.h>

// ---------------------------------------------------------------------------
// DD-LMS equalizer scan for MI455X (gfx1250, wave32).
// Serial recurrence over ~100k steps; one wave, lane l == tap t.
// CDNA5 path used: GLOBAL_LOAD_ASYNC_TO_LDS_B64 prefetch ring + S_WAIT_ASYNCCNT.
// Exact-constant simplifications: s==1, fshat==f, y==z.
// ---------------------------------------------------------------------------

#define TAPS 32
#define SPS 2
#define RING 512              // samples resident in LDS ring (16 chunks)
#define RING_MASK (RING - 1)
#define CHUNK 32              // samples per prefetch chunk (== wave32 lanes)
#define LOOKAHEAD 8           // chunks in flight
#define GROUP_STEPS 16        // steps per group (consumes exactly one chunk)

#define LR_W (1.0f / 64.0f)
#define LR_F (1.0f / 128.0f)
#define LR_B (1.0f / 2048.0f)
#define GMAX 30.0f
#define EPSV 1e-8f

// b64 async-copy builtin operates on 8-byte (2 x i32) vectors, with explicit
// global (AS1) source and LDS (AS3) destination pointers.
typedef int v2i __attribute__((ext_vector_type(2)));
typedef __attribute__((address_space(1))) v2i gv2i;  // global
typedef __attribute__((address_space(3))) v2i lv2i;  // LDS

#if defined(__has_builtin)
#if __has_builtin(__builtin_amdgcn_global_load_async_to_lds_b64)
#define HAVE_ASYNC_BUILTIN 1
#endif
#if __has_builtin(__builtin_amdgcn_s_wait_asynccnt)
#define HAVE_WAIT_BUILTIN 1
#endif
#endif
#ifndef HAVE_ASYNC_BUILTIN
#define HAVE_ASYNC_BUILTIN 0
#endif
#ifndef HAVE_WAIT_BUILTIN
#define HAVE_WAIT_BUILTIN 0
#endif

#define CFENCE() asm volatile("" ::: "memory")

#if HAVE_WAIT_BUILTIN
#define WAIT_ASYNC(n) __builtin_amdgcn_s_wait_asynccnt(n)
#else
#define WAIT_ASYNC(n) asm volatile("s_wait_asynccnt %0" ::"i"(n) : "memory")
#endif

// Copy one 32-sample chunk (re-pair 8B + im-pair 8B per sample) into the LDS
// ring asynchronously. Lane l handles sample chunk*32 + l.
__device__ __forceinline__ void issue_chunk(const float* __restrict__ xre,
                                            const float* __restrict__ xim,
                                            float4* ring, int chunk, int lane,
                                            int ntime) {
  int s = chunk * CHUNK + lane;
  if (s >= ntime) s = ntime - 1;  // clamp (harmless duplicate copy)
  const int slot = s & RING_MASK;
#if HAVE_ASYNC_BUILTIN
  __builtin_amdgcn_global_load_async_to_lds_b64(
      (gv2i*)(xre + 2 * (size_t)s), (lv2i*)&ring[slot], 0, 0);
  __builtin_amdgcn_global_load_async_to_lds_b64(
      (gv2i*)(xim + 2 * (size_t)s), (lv2i*)((float*)&ring[slot] + 2), 0, 0);
#else
  // Inline-asm fallback per CDNA5 ISA (GV mode: 64-bit vaddr, LDS addr VGPR).
  unsigned lds_lo = (unsigned)(__UINTPTR_TYPE__)(
      __attribute__((address_space(3))) void*)&ring[slot];
  unsigned lds_hi = lds_lo + 8u;
  const float* gre = xre + 2 * (size_t)s;
  const float* gim = xim + 2 * (size_t)s;
  asm volatile("global_load_async_to_lds_b64 %0, %1, off" ::"v"(lds_lo),
               "v"(gre)
               : "memory");
  asm volatile("global_load_async_to_lds_b64 %0, %1, off" ::"v"(lds_hi),
               "v"(gim)
               : "memory");
#endif
}

// Full-wave butterfly sum; every lane ends with the total (wave32).
__device__ __forceinline__ float wred(float v) {
  v += __shfl_xor(v, 1, 32);
  v += __shfl_xor(v, 2, 32);
  v += __shfl_xor(v, 4, 32);
  v += __shfl_xor(v, 8, 32);
  v += __shfl_xor(v, 16, 32);
  return v;
}

// Nearest 16-QAM level per axis: {-3,-1,1,3}/sqrt(10).
__device__ __forceinline__ float slice1(float x) {
  float t = x * 3.16227766016837933f;  // * sqrt(10)
  float r = rintf((t + 3.0f) * 0.5f);
  r = fminf(fmaxf(r, 0.0f), 3.0f);
  return (r < 1.5f) ? ((r < 0.5f) ? -0.9486832980505138f : -0.31622776601683794f)
                    : ((r < 2.5f) ? 0.31622776601683794f : 0.9486832980505138f);
}

__global__ __launch_bounds__(32) void ddlms_scan(const float* __restrict__ xre,
                                                 const float* __restrict__ xim,
                                                 float* __restrict__ out,
                                                 int n_steps, int ntime) {
  __shared__ float4 ring[RING];  // 8 KB: (re0, re1, im0, im1) per sample
  const int lane = threadIdx.x & 31;
  const int lastChunk = (ntime + CHUNK - 1) / CHUNK - 1;

  // Lane-private state: lane l holds taps w[i][j][t=l]; f/b replicated.
  float wr[2][2] = {{0.f, 0.f}, {0.f, 0.f}};
  float wi[2][2] = {{0.f, 0.f}, {0.f, 0.f}};
  float fr[2] = {1.f, 1.f}, fi[2] = {0.f, 0.f};
  float br[2] = {0.f, 0.f}, bi[2] = {0.f, 0.f};

  // Prime the pipeline.
  for (int c = 0; c < LOOKAHEAD && c <= lastChunk; ++c)
    issue_chunk(xre, xim, ring, c, lane, ntime);

  int step = 0;
  const int nGroups = (n_steps + GROUP_STEPS - 1) / GROUP_STEPS;
  for (int g = 0; g < nGroups; ++g) {
    const int c = g + LOOKAHEAD;
    if (c <= lastChunk) {
      issue_chunk(xre, xim, ring, c, lane, ntime);
      // Outstanding <= chunks (g+2..g+8) -> 14 async ops: chunks <= g+1 landed.
      WAIT_ASYNC(14);
    } else {
      WAIT_ASYNC(0);  // tail: drain everything
    }
    CFENCE();

    int send = step + GROUP_STEPS;
    if (send > n_steps) send = n_steps;
    for (; step < send; ++step) {
      const int pos = step * SPS;
      const float4 u4 = ring[(pos + lane) & RING_MASK];
      const float u0r = u4.x, u1r = u4.y, u0i = u4.z, u1i = u4.w;

      // Per-lane partials of v_i = sum_{j,t} w[i][j][t] * u[t][j].
      float p0r = wr[0][0] * u0r - wi[0][0] * u0i + wr[0][1] * u1r - wi[0][1] * u1i;
      float p0i = wr[0][0] * u0i + wi[0][0] * u0r + wr[0][1] * u1i + wi[0][1] * u1r;
      float p1r = wr[1][0] * u0r - wi[1][0] * u0i + wr[1][1] * u1r - wi[1][1] * u1i;
      float p1i = wr[1][0] * u0i + wi[1][0] * u0r + wr[1][1] * u1i + wi[1][1] * u1r;
      float pn = u0r * u0r + u0i * u0i + u1r * u1r + u1i * u1i;

      float vr[2], vi[2];
      vr[0] = wred(p0r);
      vi[0] = wred(p0i);
      vr[1] = wred(p1r);
      vi[1] = wred(p1i);
      const float un = wred(pn);
      const float inv_un = 1.0f / (un + EPSV);

      float z[4], ewr[2], ewi[2];
#pragma unroll
      for (int i = 0; i < 2; ++i) {
        const float kr = vr[i] * fr[i] - vi[i] * fi[i];
        const float ki = vr[i] * fi[i] + vi[i] * fr[i];
        const float zr = kr + br[i], zi = ki + bi[i];  // z == q (fshat==f, s==1)
        z[2 * i] = zr;
        z[2 * i + 1] = zi;
        const float dr = slice1(zr), di = slice1(zi);
        const float ebr = dr - zr, ebi = di - zi;       // e_b
        const float af = sqrtf(fr[i] * fr[i] + fi[i] * fi[i]);
        const float psr = fr[i] / af, psm = -fi[i] / af;  // psi = conj(f)/|f|
        const float dbr = dr - br[i], dbi = di - bi[i];
        ewr[i] = dbr * psr - dbi * psm - vr[i];
        ewi[i] = dbr * psm + dbi * psr - vi[i];
        const float efr = dbr - kr, efi = dbi - ki;     // e_f
        const float ivv = 1.0f / (vr[i] * vr[i] + vi[i] * vi[i] + EPSV);
        float gfr = -ivv * (efr * vr[i] + efi * vi[i]); // -e_f*conj(v)/(|v|^2+eps)
        float gfi = -ivv * (efi * vr[i] - efr * vi[i]);
        const float ag = sqrtf(gfr * gfr + gfi * gfi);
        const float sc = fminf(1.0f, GMAX / fmaxf(ag, 1e-30f));
        fr[i] -= LR_F * gfr * sc;
        fi[i] -= LR_F * gfi * sc;
        br[i] += LR_B * ebr;  // b -= LR_B * (-e_b)
        bi[i] += LR_B * ebi;
      }

      // Tap update: gw = -inv_un * e_w_i * conj(u[t][j]), element-wise clip 30.
#pragma unroll
      for (int i = 0; i < 2; ++i) {
#pragma unroll
        for (int j = 0; j < 2; ++j) {
          const float ujr = j ? u1r : u0r;
          const float uji = j ? u1i : u0i;
          const float gwr = -inv_un * (ewr[i] * ujr + ewi[i] * uji);
          const float gwi = -inv_un * (ewi[i] * ujr - ewr[i] * uji);
          const float ag = sqrtf(gwr * gwr + gwi * gwi);
          const float sc = fminf(1.0f, GMAX / fmaxf(ag, 1e-30f));
          wr[i][j] -= LR_W * gwr * sc;
          wi[i][j] -= LR_W * gwi * sc;
        }
      }

      if (lane == 0) {  // y[n] == z[n] (pre-update state), layout (n, i, re/im)
        reinterpret_cast<float4*>(out)[step] =
            make_float4(z[0], z[1], z[2], z[3]);
      }
    }
  }
}

extern "C" void kernel_launch(void* const* d_in, const int* in_sizes, int n_in,
                              void* d_out, int out_size, void* d_ws,
                              size_t ws_size, hipStream_t stream) {
  const float* xre = (const float*)d_in[0];
  const float* xim = (const float*)d_in[1];
  float* out = (float*)d_out;
  const int ntime = in_sizes[0] / 2;             // 200000
  const int n_steps = (ntime - TAPS) / SPS + 1;  // 99985
  ddlms_scan<<<dim3(1), dim3(32), 0, stream>>>(xre, xim, out, n_steps, ntime);
}